// Model_21629455302827
// MI455X (gfx1250) — compile-verified
//
#include <hip/hip_runtime.h>
#include <stdint.h>

#define TILE_WORDS 8192
#define BLOCK 256
#define CHUNK_WORDS (BLOCK * 4)              // 1024 words = 4 KB per chunk
#define CHUNKS (TILE_WORDS / CHUNK_WORDS)    // 8

__device__ __forceinline__ uint32_t fmix32(uint32_t x) {
    x ^= x >> 16;
    x *= 2246822507u;
    x ^= x >> 13;
    x *= 3266489909u;
    x ^= x >> 16;
    return x;
}

__device__ __forceinline__ uint32_t rotl32(uint32_t x, int r) {
    return (x << r) | (x >> (32 - r));   // -> v_alignbit_b32
}

// CDNA5 async global->LDS copy, 16B per lane, tracked by ASYNCcnt.
// Stream-once data: TH_LOAD_NT keeps the 256MB stream from thrashing the 192MB L2.
// Generic pointers to __shared__ carry the LDS byte offset in their low 32 bits
// so truncation yields the DS address.
__device__ __forceinline__ void async_copy_b128(uint32_t lds_off, const uint32_t* gaddr) {
    asm volatile("global_load_async_to_lds_b128 %0, %1, off th:TH_LOAD_NT"
                 :: "v"(lds_off), "v"(gaddr)
                 : "memory");
}

__global__ __launch_bounds__(BLOCK) void hash_tiles_kernel(
        const uint32_t* __restrict__ in,
        unsigned long long* __restrict__ out,
        uint32_t nbytes) {
    __shared__ uint4 lbuf[2][BLOCK];             // 8 KB double buffer (wave-private slices)
    __shared__ uint32_t ws1[BLOCK / 32], ws2[BLOCK / 32];

    const uint32_t tile = blockIdx.x;
    const uint32_t t    = threadIdx.x;
    const uint32_t base = tile * (uint32_t)TILE_WORDS + t * 4u;   // first word index for this thread
    const uint32_t* g   = in + base;

    const uint32_t POSA = 668265261u;   // POS_C1
    const uint32_t POSB = 374761393u;   // POS_C2
    const uint32_t S1   = 608135816u;   // SEED
    const uint32_t S2   = (uint32_t)(608135816u * 2654435761u) ^ 3735928559u;

    const uint32_t loff0 = (uint32_t)(uintptr_t)&lbuf[0][t];
    const uint32_t loff1 = (uint32_t)(uintptr_t)&lbuf[1][t];

    // Prime the pipeline: chunk 0 -> buffer 0.
    async_copy_b128(loff0, g);

    // Strength-reduced index streams: mA = idx*POSA + S1, mB = idx*POSB + S2.
    uint32_t idx = base;
    uint32_t mA  = base * POSA + S1;
    uint32_t mB  = base * POSB + S2;
    uint32_t a1 = 0u, a2 = 0u;

#pragma unroll
    for (int j = 0; j < CHUNKS; ++j) {
        if (j + 1 < CHUNKS) {
            // Prefetch next chunk into the other buffer while chunk j lands.
            async_copy_b128(((j + 1) & 1) ? loff1 : loff0, g + (j + 1) * CHUNK_WORDS);
            asm volatile("s_wait_asynccnt 1" ::: "memory");   // chunk j complete (in-order)
        } else {
            asm volatile("s_wait_asynccnt 0" ::: "memory");
        }

        const uint4 v = lbuf[j & 1][t];                       // ds_load_b128
        const uint32_t wv[4] = {v.x, v.y, v.z, v.w};

#pragma unroll
        for (int c = 0; c < 4; ++c) {
            const uint32_t iu = idx + (uint32_t)c;
            const uint32_t p1 = (mA + (uint32_t)c * POSA) ^ rotl32(iu, 15);
            const uint32_t p2 = (mB + (uint32_t)c * POSB) ^ rotl32(iu, 13);
            a1 += fmix32(wv[c] ^ p1);
            a2 += fmix32(wv[c] ^ p2);
        }

        idx += (uint32_t)CHUNK_WORDS;
        mA  += (uint32_t)CHUNK_WORDS * POSA;
        mB  += (uint32_t)CHUNK_WORDS * POSB;
    }

    // Intra-wave butterfly reduction (wave32).
#pragma unroll
    for (int off = 16; off > 0; off >>= 1) {
        a1 += (uint32_t)__shfl_xor((int)a1, off, 32);
        a2 += (uint32_t)__shfl_xor((int)a2, off, 32);
    }

    if ((t & 31u) == 0u) {
        ws1[t >> 5] = a1;
        ws2[t >> 5] = a2;
    }
    __syncthreads();

    if (t == 0u) {
        uint32_t h1 = 0u, h2 = 0u;
#pragma unroll
        for (int w = 0; w < BLOCK / 32; ++w) {
            h1 += ws1[w];
            h2 += ws2[w];
        }
        h1 = fmix32(h1 ^ nbytes);
        h2 = fmix32(h2 ^ nbytes);
        out[tile] = ((unsigned long long)h1 << 32) | (unsigned long long)h2;
    }
}

extern "C" void kernel_launch(void* const* d_in, const int* in_sizes, int n_in,
                              void* d_out, int out_size, void* d_ws, size_t ws_size,
                              hipStream_t stream) {
    (void)n_in; (void)d_ws; (void)ws_size; (void)out_size;

    const uint32_t* in = (const uint32_t*)d_in[0];          // int32 input, bit-identical as uint32
    unsigned long long* out = (unsigned long long*)d_out;   // reference output dtype: int64

    const int n = in_sizes[0];                              // 67108864, exact multiple of TILE
    const int tiles = n / TILE_WORDS;                       // 8192
    const uint32_t nbytes = (uint32_t)((unsigned long long)n * 4ull);

    hipLaunchKernelGGL(hash_tiles_kernel, dim3(tiles), dim3(BLOCK), 0, stream,
                       in, out, nbytes);
}